// GCN_9723805958218
// MI455X (gfx1250) — compile-verified
//
#include <hip/hip_runtime.h>

// ---------------------------------------------------------------------------
// GCN forward for MI455X (gfx1250, wave32).
// GEMMs use V_WMMA_F32_16X16X4_F32 (fp32 in, fp32 acc — GEMM cost is ~10
// GFLOP, trivially compute-bound-free; the real cost is the ~2.5 GB of edge
// gather/scatter traffic, which we service with coalesced float4 gathers and
// fp32 global atomics that land in the 192 MB L2).
// ---------------------------------------------------------------------------

typedef float v2f __attribute__((ext_vector_type(2)));
typedef float v8f __attribute__((ext_vector_type(8)));

// --------------------------- small utility kernels -------------------------

__global__ void zero_f32(float* __restrict__ p, long long n) {
  long long i = (long long)blockIdx.x * blockDim.x + threadIdx.x;
  long long stride = (long long)gridDim.x * blockDim.x;
  for (; i < n; i += stride) p[i] = 0.0f;
}

__global__ void deg_init(float* __restrict__ deg, int n) {
  int i = blockIdx.x * blockDim.x + threadIdx.x;
  if (i < n) deg[i] = 1.0f;           // self-loop
}

__global__ void deg_accum(float* __restrict__ deg, const int* __restrict__ dst, int E) {
  int e = blockIdx.x * blockDim.x + threadIdx.x;
  if (e < E) atomicAdd(&deg[dst[e]], 1.0f);
}

__global__ void dinv_kernel(float* __restrict__ dinv, const float* __restrict__ deg, int n) {
  int i = blockIdx.x * blockDim.x + threadIdx.x;
  if (i < n) dinv[i] = rsqrtf(deg[i]);
}

// ------------------------------ WMMA GEMM ----------------------------------
// C[M x NOUT] = A[M x 128] @ B[128 x NOUT], fp32, row-major.
// One wave per 16-row strip. A fragments (16x4 per k-step) are register-
// cached across all column tiles; B fragments reloaded per tile (hot in L0).
// ISA layout (16x16x4 f32):
//   A: lanes 0-15 -> M=lane, k-pair {k, k+1}; lanes 16-31 -> {k+2, k+3}
//   B: lanes 0-15 -> N=lane, rows {k, k+1}; lanes 16-31 -> rows {k+2, k+3}
//   C/D: vgpr r -> row (r + 8*(lane>>4)), col (lane&15)

template <int NOUT>
__global__ void gemm_k128_wmma(const float* __restrict__ A,
                               const float* __restrict__ B,
                               float* __restrict__ C,
                               int M) {
  constexpr int K = 128;
  const int wave = (blockIdx.x * blockDim.x + threadIdx.x) >> 5;
  const int lane = threadIdx.x & 31;
  const int row0 = wave * 16;
  if (row0 >= M) return;                 // whole-wave exit: EXEC stays all-1s

  const int lhalf = lane >> 4;           // 0: k,k+1   1: k+2,k+3
  const int lmod  = lane & 15;

  int rowA = row0 + lmod;
  if (rowA > M - 1) rowA = M - 1;        // clamp (guard is on the store side)
  const float* arow = A + (long long)rowA * K + 2 * lhalf;

  // Register-cache all 32 A fragments (64 VGPRs).
  v2f afrag[K / 4];
#pragma unroll
  for (int s = 0; s < K / 4; ++s) {
    afrag[s].x = arow[4 * s + 0];
    afrag[s].y = arow[4 * s + 1];
  }

  for (int ct = 0; ct < NOUT / 16; ++ct) {
    const int col0 = ct * 16;
    v8f acc = {};
#pragma unroll
    for (int s = 0; s < K / 4; ++s) {
      const int k = 4 * s + 2 * lhalf;
      v2f b;
      b.x = B[(long long)(k + 0) * NOUT + col0 + lmod];
      b.y = B[(long long)(k + 1) * NOUT + col0 + lmod];
      acc = __builtin_amdgcn_wmma_f32_16x16x4_f32(
          /*neg_a=*/false, afrag[s], /*neg_b=*/false, b,
          /*c_mod=*/(short)0, acc, /*reuse_a=*/false, /*reuse_b=*/false);
    }
#pragma unroll
    for (int r = 0; r < 8; ++r) {
      const int row = row0 + r + 8 * lhalf;
      if (row < M) C[(long long)row * NOUT + col0 + lmod] = acc[r];
    }
  }
}

// ----------------------------- edge scatter --------------------------------
// F=128: one wave per edge, each lane gathers float4 and issues 4 fp32
// atomics (targets fit in 192MB L2).
__global__ void scatter_f128(const float* __restrict__ H,
                             const int* __restrict__ src,
                             const int* __restrict__ dst,
                             const float* __restrict__ dinv,
                             float* __restrict__ agg, int E) {
  long long t = (long long)blockIdx.x * blockDim.x + threadIdx.x;
  int e = (int)(t >> 5);
  if (e >= E) return;
  int lane = (int)(t & 31);
  int s = src[e], d = dst[e];
  float coef = dinv[s] * dinv[d];
  float4 v = ((const float4*)(H + (long long)s * 128))[lane];
  float* ap = agg + (long long)d * 128 + lane * 4;
  atomicAdd(ap + 0, v.x * coef);
  atomicAdd(ap + 1, v.y * coef);
  atomicAdd(ap + 2, v.z * coef);
  atomicAdd(ap + 3, v.w * coef);
}

// F=64: two edges per wave (16 lanes each, float4 per lane).
__global__ void scatter_f64(const float* __restrict__ H,
                            const int* __restrict__ src,
                            const int* __restrict__ dst,
                            const float* __restrict__ dinv,
                            float* __restrict__ agg, int E) {
  long long t = (long long)blockIdx.x * blockDim.x + threadIdx.x;
  int e = (int)(t >> 4);
  if (e >= E) return;
  int li = (int)(t & 15);
  int s = src[e], d = dst[e];
  float coef = dinv[s] * dinv[d];
  float4 v = ((const float4*)(H + (long long)s * 64))[li];
  float* ap = agg + (long long)d * 64 + li * 4;
  atomicAdd(ap + 0, v.x * coef);
  atomicAdd(ap + 1, v.y * coef);
  atomicAdd(ap + 2, v.z * coef);
  atomicAdd(ap + 3, v.w * coef);
}

// ------------------------------ combines -----------------------------------
// h = relu(agg + h0*dinv^2 + b0 + l0 + bl0), written in place over agg.
__global__ void combine0(float* __restrict__ agg,
                         const float* __restrict__ h0,
                         const float* __restrict__ l0,
                         const float* __restrict__ b0,
                         const float* __restrict__ bl0,
                         const float* __restrict__ dinv, int Nn) {
  long long t = (long long)blockIdx.x * blockDim.x + threadIdx.x;
  long long total = (long long)Nn * 128;
  if (t >= total) return;
  int node = (int)(t >> 7);
  int f = (int)(t & 127);
  float di = dinv[node];
  float v = agg[t] + h0[t] * di * di + b0[f] + l0[t] + bl0[f];
  agg[t] = fmaxf(v, 0.0f);
}

// out = agg + h1*dinv^2 + b1 + l1 + bl1  (no relu)
__global__ void combine1(float* __restrict__ out,
                         const float* __restrict__ agg,
                         const float* __restrict__ h1,
                         const float* __restrict__ l1,
                         const float* __restrict__ b1,
                         const float* __restrict__ bl1,
                         const float* __restrict__ dinv, int Nn) {
  long long t = (long long)blockIdx.x * blockDim.x + threadIdx.x;
  long long total = (long long)Nn * 64;
  if (t >= total) return;
  int node = (int)(t >> 6);
  int f = (int)(t & 63);
  float di = dinv[node];
  out[t] = agg[t] + h1[t] * di * di + b1[f] + l1[t] + bl1[f];
}

// ------------------------------- driver ------------------------------------

static inline int cdivll(long long a, long long b) { return (int)((a + b - 1) / b); }

extern "C" void kernel_launch(void* const* d_in, const int* in_sizes, int n_in,
                              void* d_out, int out_size, void* d_ws, size_t ws_size,
                              hipStream_t stream) {
  const float* x   = (const float*)d_in[0];
  const int*   ei  = (const int*)d_in[1];
  const float* W0  = (const float*)d_in[2];
  const float* b0  = (const float*)d_in[3];
  const float* Wl0 = (const float*)d_in[4];
  const float* bl0 = (const float*)d_in[5];
  const float* W1  = (const float*)d_in[6];
  const float* b1  = (const float*)d_in[7];
  const float* Wl1 = (const float*)d_in[8];
  const float* bl1 = (const float*)d_in[9];

  const int N = in_sizes[0] / 128;
  const int E = in_sizes[1] / 2;
  const int* src = ei;
  const int* dst = ei + E;

  // workspace layout (floats): deg[N] | dinv[N] | bufA[N*128] | bufB[N*128] | bufC[N*128]
  float* deg  = (float*)d_ws;
  float* dinv = deg + N;
  float* bufA = dinv + N;                       // h0, then h1
  float* bufB = bufA + (long long)N * 128;      // l0, then l1
  float* bufC = bufB + (long long)N * 128;      // agg0 -> h (in place), then agg1
  float* out  = (float*)d_out;

  const int T = 256;  // 8 wave32s per block

  // ---- degree / normalization ----
  deg_init<<<cdivll(N, T), T, 0, stream>>>(deg, N);
  deg_accum<<<cdivll(E, T), T, 0, stream>>>(deg, dst, E);
  dinv_kernel<<<cdivll(N, T), T, 0, stream>>>(dinv, deg, N);

  // ---- layer 0 GEMMs: h0 = x@W0, l0 = x@Wl0 ----
  const long long gthr = (long long)cdivll(N, 16) * 32;
  gemm_k128_wmma<128><<<cdivll(gthr, T), T, 0, stream>>>(x, W0, bufA, N);
  gemm_k128_wmma<128><<<cdivll(gthr, T), T, 0, stream>>>(x, Wl0, bufB, N);

  // ---- layer 0 aggregate + combine ----
  zero_f32<<<2048, T, 0, stream>>>(bufC, (long long)N * 128);
  scatter_f128<<<cdivll((long long)E * 32, T), T, 0, stream>>>(bufA, src, dst, dinv, bufC, E);
  combine0<<<cdivll((long long)N * 128, T), T, 0, stream>>>(bufC, bufA, bufB, b0, bl0, dinv, N);

  // ---- layer 1 GEMMs: h1 = h@W1, l1 = h@Wl1 (h lives in bufC) ----
  gemm_k128_wmma<64><<<cdivll(gthr, T), T, 0, stream>>>(bufC, W1, bufA, N);
  gemm_k128_wmma<64><<<cdivll(gthr, T), T, 0, stream>>>(bufC, Wl1, bufB, N);

  // ---- layer 1 aggregate + combine (agg1 reuses bufC; h no longer needed) ----
  zero_f32<<<2048, T, 0, stream>>>(bufC, (long long)N * 64);
  scatter_f64<<<cdivll((long long)E * 16, T), T, 0, stream>>>(bufA, src, dst, dinv, bufC, E);
  combine1<<<cdivll((long long)N * 64, T), T, 0, stream>>>(out, bufC, bufA, bufB, b1, bl1, dinv, N);
}